// DynamicRotatedROICropper_86689619903165
// MI455X (gfx1250) — compile-verified
//
#include <hip/hip_runtime.h>
#include <math.h>

// Problem constants (static shapes from the reference).
#define BATCH 4
#define NBOX  512          // B*N = 4*128
#define CCH   3
#define HI    800
#define WI    1216
#define HOUT  48
#define WMAX  320
#define HW    (HI * WI)          // 972800
#define CHW   (CCH * HW)         // 2918400
#define OHW   (HOUT * WMAX)      // 15360
#define OCHW  (CCH * OHW)        // 46080

// -------------------------------------------------------------------------
// Kernel 1: per-box parameter prep (512 threads total).
// Params per box (8 floats): p1x, p1y, (p2-p1).x, (p2-p1).y, (p0-p1).x,
// (p0-p1).y, rw, pad.  Matches reference math (rintf == round-half-even).
// -------------------------------------------------------------------------
__global__ __launch_bounds__(256)
void rroi_prep_kernel(const float* __restrict__ boxes, float* __restrict__ params)
{
    int nb = blockIdx.x * blockDim.x + threadIdx.x;
    if (nb >= NBOX) return;

    const float cx  = boxes[nb * 5 + 0];
    const float cy  = boxes[nb * 5 + 1];
    const float w   = boxes[nb * 5 + 2];
    const float h   = boxes[nb * 5 + 3];
    const float ang = boxes[nb * 5 + 4];

    // rw = clip(round_half_even(72*w/h), 16, 320)
    float rw = rintf(72.0f * w / h);
    rw = fminf(fmaxf(rw, 16.0f), 320.0f);

    const float ar = -ang * 0.017453292519943295f;   // deg2rad(-ang)
    const float cb = cosf(ar) * 0.5f;
    const float sa = sinf(ar) * 0.5f;

    const float p0x = cx - sa * h - cb * w;
    const float p0y = cy + cb * h - sa * w;
    const float p1x = cx + sa * h - cb * w;
    const float p1y = cy - cb * h - sa * w;
    const float p2x = cx + sa * h + cb * w;
    const float p2y = cy - cb * h + sa * w;

    float* o = params + nb * 8;
    o[0] = p1x;
    o[1] = p1y;
    o[2] = p2x - p1x;
    o[3] = p2y - p1y;
    o[4] = p0x - p1x;
    o[5] = p0y - p1y;
    o[6] = rw;
    o[7] = 0.0f;
}

// -------------------------------------------------------------------------
// Kernel 2: bilinear gather + blend.
// grid = (NBOX, HOUT), block = WMAX (= 320 threads = 10 wave32s).
// 12 gather loads (L2-resident input) + 3 coalesced NON-TEMPORAL stores
// (output is write-once; keep it out of L2 so the input stays resident).
// -------------------------------------------------------------------------
__global__ __launch_bounds__(WMAX)
void rroi_crop_kernel(const float* __restrict__ x,
                      const float* __restrict__ params,
                      float* __restrict__ out)
{
    const int u  = threadIdx.x;   // output column 0..319
    const int v  = blockIdx.y;    // output row    0..47
    const int nb = blockIdx.x;    // box           0..511

    // Block-uniform params -> scalar loads.
    const float* P = params + nb * 8;
    const float p1x = P[0], p1y = P[1];
    const float d2x = P[2], d2y = P[3];
    const float d0x = P[4], d0y = P[5];
    const float rw  = P[6];

    // Same association as reference: fu = u/rw, fv = v/48.
    const float fu = (float)u / rw;
    const float fv = (float)v / (float)HOUT;
    const float sx = p1x + fu * d2x + fv * d0x;
    const float sy = p1y + fu * d2y + fv * d0y;

    const float x0f = floorf(sx);
    const float y0f = floorf(sy);
    const float wx  = sx - x0f;
    const float wy  = sy - y0f;
    const int   xi  = (int)x0f;
    const int   yi  = (int)y0f;
    const int   x1  = xi + 1;
    const int   y1  = yi + 1;

    // Zero-border validity + column padding, folded into the 4 tap weights.
    const float pad = ((float)u < rw) ? 1.0f : 0.0f;
    const bool vx0 = (xi >= 0) & (xi < WI);
    const bool vx1 = (x1 >= 0) & (x1 < WI);
    const bool vy0 = (yi >= 0) & (yi < HI);
    const bool vy1 = (y1 >= 0) & (y1 < HI);

    const float w00 = (1.0f - wy) * (1.0f - wx) * ((vx0 & vy0) ? pad : 0.0f);
    const float w01 = (1.0f - wy) * wx          * ((vx1 & vy0) ? pad : 0.0f);
    const float w10 = wy * (1.0f - wx)          * ((vx0 & vy1) ? pad : 0.0f);
    const float w11 = wy * wx                   * ((vx1 & vy1) ? pad : 0.0f);

    // Clamped tap addresses (loads stay in-bounds; zero weight kills value).
    const int xc0 = min(max(xi, 0), WI - 1);
    const int xc1 = min(max(x1, 0), WI - 1);
    const int yc0 = min(max(yi, 0), HI - 1);
    const int yc1 = min(max(y1, 0), HI - 1);

    const int i00 = yc0 * WI + xc0;
    const int i01 = yc0 * WI + xc1;
    const int i10 = yc1 * WI + xc0;
    const int i11 = yc1 * WI + xc1;

    const float* base = x + (size_t)(nb >> 7) * CHW;  // bidx = nb / 128
    float* po = out + (size_t)nb * OCHW + (size_t)v * WMAX + u;

    // gfx1250 prefetch: one line per *next* channel plane (3.9 MB apart,
    // cold lines) issued before the plane-0 gathers retire.  Exactly one
    // prefetch per future plane -- no duplicates.
    __builtin_prefetch(base + 1 * HW + i00, 0, 0);
    __builtin_prefetch(base + 2 * HW + i00, 0, 0);

#pragma unroll
    for (int c = 0; c < CCH; ++c) {
        const float* p = base + c * HW;
        const float t00 = p[i00];
        const float t01 = p[i01];
        const float t10 = p[i10];
        const float t11 = p[i11];
        const float val = t00 * w00 + t01 * w01 + t10 * w10 + t11 * w11;
        // NT store: stream write-once output past L2, keep input resident.
        __builtin_nontemporal_store(val, po + c * OHW);
    }
}

// -------------------------------------------------------------------------
// Launch: prep (512 boxes) then crop (512 x 48 blocks of 320 threads).
// -------------------------------------------------------------------------
extern "C" void kernel_launch(void* const* d_in, const int* in_sizes, int n_in,
                              void* d_out, int out_size, void* d_ws, size_t ws_size,
                              hipStream_t stream)
{
    const float* x     = (const float*)d_in[0];   // [4,3,800,1216] f32
    const float* boxes = (const float*)d_in[1];   // [4,128,5] f32
    float* out         = (float*)d_out;           // [512,3,48,320] f32
    float* params      = (float*)d_ws;            // 512*8 floats = 16 KB

    rroi_prep_kernel<<<dim3((NBOX + 255) / 256), dim3(256), 0, stream>>>(boxes, params);

    dim3 grid(NBOX, HOUT);
    rroi_crop_kernel<<<grid, dim3(WMAX), 0, stream>>>(x, params, out);
}